// SelfGradDynamicMLP2L_46488726012472
// MI455X (gfx1250) — compile-verified
//
#include <hip/hip_runtime.h>
#include <math.h>

#define LRATE 0.001f
#define EPSN  1e-12f

typedef float v2f __attribute__((ext_vector_type(2)));
typedef float v4f __attribute__((ext_vector_type(4)));
typedef float v8f __attribute__((ext_vector_type(8)));

static constexpr int B_  = 64;
static constexpr int IN_ = 2048;
static constexpr int HID_ = 512;
static constexpr int OUT_ = 128;
static constexpr int QCH = 8, QLEN = IN_ / QCH;   // 256
static constexpr int HCH = 8, HLEN = HID_ / HCH;  // 64

// ---------------------------------------------------------------------------
// fp32 WMMA GEMM: C[M,N] = A[M,K] @ B[K,N]  (TRANSB: B stored [N,K])
// One wave32 per 16x32 C tile (two 16x16 subtiles share the A fragment).
// V_WMMA_F32_16X16X4_F32 along K; TRANSB is a compile-time template arg so
// the inner loop is branch-free (pointer strides only).
// A-frag (16x4 f32): lane L holds A[m=L%16][kbase + j], kbase=(L/16)*2, j=0,1
// B-frag (4x16 f32): lane L holds B[kbase + j][n=L%16]
// C/D   (16x16 f32): VGPR r: lanes 0-15 -> M=r, lanes 16-31 -> M=r+8; N=L%16
// ---------------------------------------------------------------------------
template <int TRANSB>
__global__ void wmma_gemm_f32(const float* __restrict__ A,
                              const float* __restrict__ B,
                              float* __restrict__ C,
                              int M, int N, int K,
                              int lda, int ldb, int ldc)
{
    const int lane = threadIdx.x & 31;
    const int wave = threadIdx.x >> 5;
    const int ntn  = N >> 5;                          // 32-wide N tiles
    const int tile = blockIdx.x * (blockDim.x >> 5) + wave;
    if (tile >= (M >> 4) * ntn) return;               // whole-wave exit: EXEC all-ones
    const int mt = tile / ntn;
    const int nt = tile % ntn;

    const int mrow = lane & 15;
    const int kh   = (lane >> 4) << 1;                // 0 or 2
    const int n0   = nt * 32 + (lane & 15);
    const int n1   = n0 + 16;

    const float* Arow = A + (size_t)(mt * 16 + mrow) * lda + kh;
    const float* B0;
    const float* B1;
    if (TRANSB) {                                      // B stored [N,K], k contiguous
        B0 = B + (size_t)n0 * ldb + kh;
        B1 = B + (size_t)n1 * ldb + kh;
    } else {                                           // B stored [K,N]
        B0 = B + (size_t)kh * ldb + n0;
        B1 = B0 + 16;
    }

    v8f acc0 = {};
    v8f acc1 = {};
#pragma unroll 2
    for (int k0 = 0; k0 < K; k0 += 4) {
        v2f a, b0, b1;
        a.x = Arow[0];
        a.y = Arow[1];
        if (TRANSB) {
            b0.x = B0[0]; b0.y = B0[1];
            b1.x = B1[0]; b1.y = B1[1];
            B0 += 4; B1 += 4;
        } else {
            b0.x = B0[0]; b0.y = B0[ldb];
            b1.x = B1[0]; b1.y = B1[ldb];
            B0 += (size_t)ldb * 4; B1 += (size_t)ldb * 4;
        }
        Arow += 4;
        acc0 = __builtin_amdgcn_wmma_f32_16x16x4_f32(
                   false, a, false, b0, (short)0, acc0, false, false);
        acc1 = __builtin_amdgcn_wmma_f32_16x16x4_f32(
                   false, a, false, b1, (short)0, acc1, false, false);
    }

    const int row0 = mt * 16 + ((lane >> 4) << 3);
#pragma unroll
    for (int r = 0; r < 8; ++r) {
        C[(size_t)(row0 + r) * ldc + n0] = acc0[r];
        C[(size_t)(row0 + r) * ldc + n1] = acc1[r];
    }
}

// ---------------------------------------------------------------------------
// h_part[b][c][o] = sum_{q in chunk c} x[b,q] * dW0[b,q,o]
// 128 threads, each owns 4 contiguous o's; b128 NT loads stream 256 MB of dW0.
// ---------------------------------------------------------------------------
__global__ void k_stream_dot1(const float* __restrict__ x,
                              const float* __restrict__ dW0,
                              float* __restrict__ h_part)
{
    const int b = blockIdx.x, c = blockIdx.y;
    const int o4 = threadIdx.x << 2;                  // 128 threads -> o in [0,512)
    const float* xb = x + b * IN_ + c * QLEN;          // wave-uniform (scalar) loads
    const float* w  = dW0 + (size_t)(b * IN_ + c * QLEN) * HID_ + o4;
    v4f acc = {};
#pragma unroll 4
    for (int q = 0; q < QLEN; ++q) {
        const v4f v = __builtin_nontemporal_load((const v4f*)w);
        const float xv = xb[q];
        acc.x = fmaf(xv, v.x, acc.x);
        acc.y = fmaf(xv, v.y, acc.y);
        acc.z = fmaf(xv, v.z, acc.z);
        acc.w = fmaf(xv, v.w, acc.w);
        w += HID_;
    }
    *(v4f*)(h_part + (size_t)(b * QCH + c) * HID_ + o4) = acc;
}

__global__ void k_finish1(const float* __restrict__ h_shared,
                          const float* __restrict__ h_part,
                          const float* __restrict__ bias0,
                          const float* __restrict__ dbias0,
                          float* __restrict__ tanh_out)
{
    const int b = blockIdx.x, o = threadIdx.x;
    float s = h_shared[b * HID_ + o] + bias0[o] + dbias0[b * HID_ + o];
#pragma unroll
    for (int c = 0; c < QCH; ++c) s += h_part[(b * QCH + c) * HID_ + o];
    tanh_out[b * HID_ + o] = tanhf(s);
}

// ---------------------------------------------------------------------------
// out2_part[b][c][o] = sum_{h in chunk c} tanh_out[b,h] * dW1[b,h,o]
// ---------------------------------------------------------------------------
__global__ void k_stream_dot2(const float* __restrict__ tanh_out,
                              const float* __restrict__ dW1,
                              float* __restrict__ o2_part)
{
    const int b = blockIdx.x, c = blockIdx.y, o = threadIdx.x;   // 128 threads
    const float* tb = tanh_out + b * HID_ + c * HLEN;
    const float* w  = dW1 + (size_t)(b * HID_ + c * HLEN) * OUT_ + o;
    float acc = 0.f;
#pragma unroll 4
    for (int h = 0; h < HLEN; ++h) {
        acc = fmaf(tb[h], __builtin_nontemporal_load(w), acc);
        w += OUT_;
    }
    o2_part[(size_t)(b * HCH + c) * OUT_ + o] = acc;
}

// ---------------------------------------------------------------------------
// Per-sample normalize + analytic g_b2 (LDS block reductions over 128 lanes)
// ---------------------------------------------------------------------------
__device__ __forceinline__ float block_reduce128(float v, float* red)
{
    const int t = threadIdx.x;
    red[t] = v;
    __syncthreads();
#pragma unroll
    for (int s = 64; s > 0; s >>= 1) {
        if (t < s) red[t] += red[t + s];
        __syncthreads();
    }
    const float r = red[0];
    __syncthreads();
    return r;
}

__global__ void k_finish2(const float* __restrict__ o2_shared,
                          const float* __restrict__ o2_part,
                          const float* __restrict__ bias1,
                          const float* __restrict__ dbias1,
                          const float* __restrict__ ut,
                          float* __restrict__ g_b2,
                          float* __restrict__ out_l2,
                          float* __restrict__ out_ndb1)
{
    __shared__ float red[OUT_];
    const int b = blockIdx.x, o = threadIdx.x;
    float out2 = o2_shared[b * OUT_ + o] + bias1[o] + dbias1[b * OUT_ + o];
#pragma unroll
    for (int c = 0; c < HCH; ++c) out2 += o2_part[(b * HCH + c) * OUT_ + o];

    const float l2n = block_reduce128(out2 * out2, red);     // squared norm (as in ref)
    const float u   = ut[b * OUT_ + o];
    const float tn  = block_reduce128(u * u, red);
    const float t   = u / fmaxf(sqrtf(tn), EPSN);
    const float dt  = block_reduce128(t * out2, red);

    const float inv = 1.f / l2n;
    const float g   = -t * inv + (dt * inv * inv) * 2.f * out2;

    out_l2[b * OUT_ + o]   = out2 * inv;
    out_ndb1[b * OUT_ + o] = dbias1[b * OUT_ + o] - LRATE * g;
    g_b2[b * OUT_ + o]     = g;
}

// ---------------------------------------------------------------------------
// Fused: new_dW1 = dW1 - LR * tanh_out[b,h]*g_b2[b,:]  AND  row_dot[b,h] =
// dot(dW1[b,h,:], g_b2[b,:]).  One wave per (b,h) row, float4 NT stream.
// ---------------------------------------------------------------------------
__global__ void k_fuse_dw1(const float* __restrict__ dW1,
                           const float* __restrict__ g_b2,
                           const float* __restrict__ tanh_out,
                           float* __restrict__ out_ndw1,
                           float* __restrict__ row_dot)
{
    const int lane = threadIdx.x & 31;
    const int row  = blockIdx.x * (blockDim.x >> 5) + (threadIdx.x >> 5); // b*HID_+h
    const int b    = row >> 9;

    const size_t off = (size_t)row * OUT_ + lane * 4;
    v4f v = __builtin_nontemporal_load((const v4f*)(dW1 + off));
    v4f g = *(const v4f*)(g_b2 + b * OUT_ + lane * 4);

    float p = v.x * g.x + v.y * g.y + v.z * g.z + v.w * g.w;
    p += __shfl_xor(p, 16);
    p += __shfl_xor(p, 8);
    p += __shfl_xor(p, 4);
    p += __shfl_xor(p, 2);
    p += __shfl_xor(p, 1);

    const float s = LRATE * tanh_out[row];
    v4f o;
    o.x = v.x - s * g.x; o.y = v.y - s * g.y;
    o.z = v.z - s * g.z; o.w = v.w - s * g.w;
    __builtin_nontemporal_store(o, (v4f*)(out_ndw1 + off));
    if (lane == 0) row_dot[row] = p;
}

__global__ void k_finish_gb(const float* __restrict__ gbT_shared,
                            const float* __restrict__ row_dot,
                            const float* __restrict__ tanh_out,
                            const float* __restrict__ dbias0,
                            float* __restrict__ g_b,
                            float* __restrict__ out_ndb0)
{
    const int i = blockIdx.x * HID_ + threadIdx.x;
    const float th = tanh_out[i];
    const float g  = (1.f - th * th) * (gbT_shared[i] + row_dot[i]);
    g_b[i]      = g;
    out_ndb0[i] = dbias0[i] - LRATE * g;
}

// ---------------------------------------------------------------------------
// new_dW0 = dW0 - LR * x[b,q] * g_b[b,:]  — 512 MB stream, float4 NT
// ---------------------------------------------------------------------------
__global__ void k_update_dw0(const float* __restrict__ dW0,
                             const float* __restrict__ x,
                             const float* __restrict__ g_b,
                             float* __restrict__ out)
{
    const int gid = blockIdx.x * blockDim.x + threadIdx.x;   // 16M float4s
    const size_t idx = (size_t)gid * 4;
    const int rem = gid & 262143;          // (2048*512/4)-1 per sample
    const int b   = gid >> 18;
    const int q   = rem >> 7;              // 128 float4s per q-row (wave-uniform)
    const int o4  = (rem & 127) << 2;

    v4f v = __builtin_nontemporal_load((const v4f*)(dW0 + idx));
    const float s = LRATE * x[b * IN_ + q];
    v4f g = *(const v4f*)(g_b + b * HID_ + o4);
    v4f o;
    o.x = v.x - s * g.x; o.y = v.y - s * g.y;
    o.z = v.z - s * g.z; o.w = v.w - s * g.w;
    __builtin_nontemporal_store(o, (v4f*)(out + idx));
}

// ---------------------------------------------------------------------------
extern "C" void kernel_launch(void* const* d_in, const int* in_sizes, int n_in,
                              void* d_out, int out_size, void* d_ws, size_t ws_size,
                              hipStream_t stream)
{
    const float* x   = (const float*)d_in[0];
    const float* ut  = (const float*)d_in[1];
    const float* W0  = (const float*)d_in[2];
    const float* b0  = (const float*)d_in[3];
    const float* W1  = (const float*)d_in[4];
    const float* b1  = (const float*)d_in[5];
    const float* dW0 = (const float*)d_in[6];
    const float* db0 = (const float*)d_in[7];
    const float* dW1 = (const float*)d_in[8];
    const float* db1 = (const float*)d_in[9];

    float* out    = (float*)d_out;
    float* o_l2   = out;                                   // [64,128]
    float* o_ndw0 = out + 8192;                            // [64,2048,512]
    float* o_ndb0 = o_ndw0 + (size_t)67108864;             // [64,512]
    float* o_ndw1 = o_ndb0 + 32768;                        // [64,512,128]
    float* o_ndb1 = o_ndw1 + (size_t)4194304;              // [64,128]

    float* ws = (float*)d_ws;
    float* h_shared  = ws;  ws += B_ * HID_;
    float* h_part    = ws;  ws += B_ * QCH * HID_;
    float* tanh_out  = ws;  ws += B_ * HID_;
    float* o2_shared = ws;  ws += B_ * OUT_;
    float* o2_part   = ws;  ws += B_ * HCH * OUT_;
    float* g_b2      = ws;  ws += B_ * OUT_;
    float* gbT       = ws;  ws += B_ * HID_;
    float* row_dot   = ws;  ws += B_ * HID_;
    float* g_b       = ws;  ws += B_ * HID_;

    // 1) shared-weight GEMM:  h_shared = x @ W0   (64 16x32 tiles, 4 waves/blk)
    wmma_gemm_f32<0><<<16, 128, 0, stream>>>(x, W0, h_shared, 64, 512, 2048,
                                             2048, 512, 512);
    // 2) per-sample diag part of layer 1 (streams 256 MB of dW0, b128 NT)
    k_stream_dot1<<<dim3(B_, QCH), 128, 0, stream>>>(x, dW0, h_part);
    // 3) combine + bias + tanh
    k_finish1<<<B_, HID_, 0, stream>>>(h_shared, h_part, b0, db0, tanh_out);
    // 4) shared-weight GEMM:  o2_shared = tanh_out @ W1   (16 tiles)
    wmma_gemm_f32<0><<<4, 128, 0, stream>>>(tanh_out, W1, o2_shared, 64, 128, 512,
                                            512, 128, 128);
    // 5) per-sample diag part of layer 2 (streams 16 MB of dW1)
    k_stream_dot2<<<dim3(B_, HCH), OUT_, 0, stream>>>(tanh_out, dW1, o2_part);
    // 6) normalize + g_b2 + outputs (l2_normed_out2, new_d_bias_1)
    k_finish2<<<B_, OUT_, 0, stream>>>(o2_shared, o2_part, b1, db1, ut,
                                       g_b2, o_l2, o_ndb1);
    // 7) shared-weight GEMM:  gbT = g_b2 @ W1^T   (64 tiles, B transposed)
    wmma_gemm_f32<1><<<16, 128, 0, stream>>>(g_b2, W1, gbT, 64, 512, 128,
                                             128, 128, 512);
    // 8) fused dW1 update + per-row dot with g_b2 (one 16 MB pass)
    k_fuse_dw1<<<(B_ * HID_) / 8, 256, 0, stream>>>(dW1, g_b2, tanh_out,
                                                    o_ndw1, row_dot);
    // 9) g_b + new_d_bias_0
    k_finish_gb<<<B_, HID_, 0, stream>>>(gbT, row_dot, tanh_out, db0, g_b, o_ndb0);
    // 10) 512 MB streaming update of dW0 (float4 NT load/store)
    k_update_dw0<<<(67108864 / 4) / 256, 256, 0, stream>>>(dW0, x, g_b, o_ndw0);
}